// MAE_25864293057262
// MI455X (gfx1250) — compile-verified
//
#include <hip/hip_runtime.h>
#include <hip/hip_bf16.h>

typedef __attribute__((ext_vector_type(2))) float v2f;
typedef __attribute__((ext_vector_type(8))) float v8f;

#define TILE_M 128
#define TILE_N 128
#define TILE_K 16
#define KPAD 18   // row stride (floats) for K-innermost LDS tiles; 18n mod 64 injective

// Stage one 128x16 A tile and 16x128 B tile into LDS, K-innermost, zero-padded.
__device__ __forceinline__ void vae_load_stage(
    const float* __restrict__ A, const float* __restrict__ B,
    float (*__restrict__ As)[KPAD], float (*__restrict__ Bs)[KPAD],
    int tid, int blockM0, int blockN0, int kk, int N, int K, int kend)
{
    #pragma unroll
    for (int e = 0; e < 8; e++) {
        int idx = tid + e * 256;       // 0..2047
        int m = idx >> 4;
        int k = idx & 15;
        int gk = kk + k;
        float v = 0.0f;
        if (gk < kend) v = A[(size_t)(blockM0 + m) * K + gk];
        As[m][k] = v;
    }
    #pragma unroll
    for (int e = 0; e < 8; e++) {
        int idx = tid + e * 256;
        int k = idx >> 7;
        int n = idx & 127;
        int gk = kk + k;
        int gn = blockN0 + n;
        float v = 0.0f;
        if (gk < kend && gn < N) v = B[(size_t)gk * N + gn];
        Bs[n][k] = v;
    }
}

// C[M,N] = act(A[M,K(range k0..k1)] @ B + bias); mode 0: relu, 1: sigmoid(relu), 2: raw partial.
// 256 threads = 8 wave32 (4 M-slices x 2 N-slices); wave tile 32x64 = 2x4 f32 WMMA accums.
__global__ __launch_bounds__(256)
void vae_gemm_bias_act(const float* __restrict__ A, const float* __restrict__ B,
                       const float* __restrict__ bias, float* __restrict__ C,
                       int M, int N, int K, int k0, int k1, int mode)
{
    __shared__ __align__(16) float As[2][TILE_M][KPAD];   // 2 x 9 KB
    __shared__ __align__(16) float Bs[2][TILE_N][KPAD];

    const int tid   = threadIdx.x;
    const int lane  = tid & 31;
    const int wave  = tid >> 5;
    const int waveM = wave & 3;            // 32-row slice
    const int waveN = wave >> 2;           // 64-col slice
    const int half  = lane >> 4;
    const int l16   = lane & 15;
    const int blockN0 = blockIdx.x * TILE_N;
    const int blockM0 = blockIdx.y * TILE_M;

    v8f zero = {};
    v8f acc[2][4];
    #pragma unroll
    for (int i = 0; i < 2; i++)
        #pragma unroll
        for (int j = 0; j < 4; j++) acc[i][j] = zero;

    vae_load_stage(A, B, As[0], Bs[0], tid, blockM0, blockN0, k0, N, K, k1);
    __syncthreads();

    for (int kk = k0; kk < k1; kk += TILE_K) {
        const int cur = ((kk - k0) >> 4) & 1;
        if (kk + TILE_K < k1)
            vae_load_stage(A, B, As[cur ^ 1], Bs[cur ^ 1], tid,
                           blockM0, blockN0, kk + TILE_K, N, K, k1);

        const float (*__restrict__ Ac)[KPAD] = As[cur];
        const float (*__restrict__ Bc)[KPAD] = Bs[cur];
        #pragma unroll
        for (int k4 = 0; k4 < TILE_K; k4 += 4) {
            const int kb = k4 + 2 * half;  // frag layout: K = vgpr + 2*(lane>=16)
            v2f afrag[2];
            #pragma unroll
            for (int ms = 0; ms < 2; ms++)
                afrag[ms] = *(const v2f*)&Ac[waveM * 32 + ms * 16 + l16][kb];
            v2f bfrag[4];
            #pragma unroll
            for (int ns = 0; ns < 4; ns++)
                bfrag[ns] = *(const v2f*)&Bc[waveN * 64 + ns * 16 + l16][kb];
            #pragma unroll
            for (int ms = 0; ms < 2; ms++)
                #pragma unroll
                for (int ns = 0; ns < 4; ns++)
                    acc[ms][ns] = __builtin_amdgcn_wmma_f32_16x16x4_f32(
                        false, afrag[ms], false, bfrag[ns],
                        (short)0, acc[ms][ns], false, false);
        }
        __syncthreads();
    }

    // Epilogue: C layout VGPR r -> row = r + 8*(lane>=16), col = lane&15.
    #pragma unroll
    for (int ms = 0; ms < 2; ms++) {
        #pragma unroll
        for (int ns = 0; ns < 4; ns++) {
            int col = blockN0 + waveN * 64 + ns * 16 + l16;
            if (col < N) {
                float bv = (mode == 2) ? 0.0f : bias[col];
                #pragma unroll
                for (int r = 0; r < 8; r++) {
                    int row = blockM0 + waveM * 32 + ms * 16 + r + 8 * half;
                    float v = acc[ms][ns][r];
                    if (mode != 2) {
                        v = fmaxf(v + bv, 0.0f);
                        if (mode == 1) v = 1.0f / (1.0f + expf(-v));
                    }
                    C[(size_t)row * N + col] = v;
                }
            }
        }
    }
}

// Fixed-order split-K combine: C = relu(sum_z part[z] + bias)
__global__ void vae_combine_relu(const float* __restrict__ part, const float* __restrict__ bias,
                                 float* __restrict__ C, int MN, int N, int S)
{
    int i = blockIdx.x * blockDim.x + threadIdx.x;
    if (i < MN) {
        float s = 0.0f;
        for (int z = 0; z < S; z++) s += part[(size_t)z * MN + i];
        C[i] = fmaxf(s + bias[i % N], 0.0f);
    }
}

// z = eps * sig + mu
__global__ void vae_reparam(const float* __restrict__ eps, const float* __restrict__ sig,
                            const float* __restrict__ mu, float* __restrict__ z, int n)
{
    int i = blockIdx.x * blockDim.x + threadIdx.x;
    if (i < n) z[i] = fmaf(eps[i], sig[i], mu[i]);
}

// Per paired row i (0..4095): dklrow, dquad, t = trace + e. 128 threads = LATENT.
__global__ __launch_bounds__(128)
void vae_stats(const float* __restrict__ mu, const float* __restrict__ sig,
               float* __restrict__ dquad, float* __restrict__ tvec,
               float* __restrict__ dklrow)
{
    __shared__ float red[5][128];
    const int i = blockIdx.x;
    const int j = threadIdx.x;
    float m1 = mu[(size_t)i * 128 + j];
    float m2 = mu[(size_t)(i + 4096) * 128 + j];
    float s1 = sig[(size_t)i * 128 + j] + 1e-8f;
    float s2 = sig[(size_t)(i + 4096) * 128 + j] + 1e-8f;
    float sm = 0.5f * (s1 + s2);
    float mm = 0.5f * (m1 + m2);
    float dkl = -1.0f - logf(1e-8f + sm * sm) + mm * mm + sm * sm;
    float df = m2 - m1;
    red[0][j] = dkl;
    red[1][j] = s1 / s2;           // trace term
    red[2][j] = df * df / s2;      // dquad term
    red[3][j] = s1;                // det1 (product)
    red[4][j] = s2;                // det2 (product)
    __syncthreads();
    for (int s = 64; s > 0; s >>= 1) {
        if (j < s) {
            red[0][j] += red[0][j + s];
            red[1][j] += red[1][j + s];
            red[2][j] += red[2][j + s];
            red[3][j] *= red[3][j + s];
            red[4][j] *= red[4][j + s];
        }
        __syncthreads();
    }
    if (j == 0) {
        dklrow[i] = 0.5f * red[0][0];
        dquad[i]  = red[2][0];
        float e = logf(1e-8f + red[4][0] / (red[3][0] + 1e-8f)) - 128.0f;
        tvec[i] = red[1][0] + e;
    }
}

// l_diverse partials: sum_{i,k} log1p(exp(-0.5*(dquad[i] + t[k])))
__global__ __launch_bounds__(256)
void vae_ldiverse(const float* __restrict__ dquad, const float* __restrict__ tvec,
                  float* __restrict__ part)
{
    __shared__ float ts[256];
    __shared__ float red[256];
    const int i = blockIdx.x * 256 + threadIdx.x;
    const float dq = dquad[i];
    float sum = 0.0f;
    for (int kt = 0; kt < 4096; kt += 256) {
        ts[threadIdx.x] = tvec[kt + threadIdx.x];
        __syncthreads();
        #pragma unroll 8
        for (int k = 0; k < 256; k++) {
            float D = 0.5f * (dq + ts[k]);
            sum += log1pf(expf(-D));
        }
        __syncthreads();
    }
    red[threadIdx.x] = sum;
    __syncthreads();
    for (int s = 128; s > 0; s >>= 1) {
        if (threadIdx.x < s) red[threadIdx.x] += red[threadIdx.x + s];
        __syncthreads();
    }
    if (threadIdx.x == 0) part[blockIdx.x] = red[0];
}

// rec partials: sum x*clog(y) + (1-x)*clog(1-y)  (negated in final kernel)
__global__ __launch_bounds__(256)
void vae_rec(const float* __restrict__ x, const float* __restrict__ y,
             float* __restrict__ part, int n)
{
    __shared__ float red[256];
    float sum = 0.0f;
    for (int i = blockIdx.x * blockDim.x + threadIdx.x; i < n;
         i += gridDim.x * blockDim.x) {
        float xv = x[i], yv = y[i];
        float l1 = fmaxf(logf(yv), -100.0f);
        float l2 = fmaxf(logf(1.0f - yv), -100.0f);
        sum += xv * l1 + (1.0f - xv) * l2;
    }
    red[threadIdx.x] = sum;
    __syncthreads();
    for (int s = 128; s > 0; s >>= 1) {
        if (threadIdx.x < s) red[threadIdx.x] += red[threadIdx.x + s];
        __syncthreads();
    }
    if (threadIdx.x == 0) part[blockIdx.x] = red[0];
}

// Deterministic sequential final combine. l_smooth is constant over i:
// D[i,k]-mean_k = 0.5*(t[k]-tbar)  ->  sum l_smooth = 4096*sqrt(0.25*SS/127).
__global__ void vae_final(const float* __restrict__ dklrow, const float* __restrict__ recpart,
                          const float* __restrict__ ldivpart, const float* __restrict__ tvec,
                          float* __restrict__ out_loss)
{
    if (threadIdx.x == 0 && blockIdx.x == 0) {
        float dkl = 0.0f;
        for (int i = 0; i < 4096; i++) dkl += dklrow[i];
        float rec = 0.0f;
        for (int i = 0; i < 1024; i++) rec += recpart[i];
        float ldv = 0.0f;
        for (int i = 0; i < 16; i++) ldv += ldivpart[i];
        float tsum = 0.0f;
        for (int i = 0; i < 4096; i++) tsum += tvec[i];
        float tbar = tsum * (1.0f / 4096.0f);
        float ss = 0.0f;
        for (int i = 0; i < 4096; i++) { float d = tvec[i] - tbar; ss += d * d; }
        float lsm = 4096.0f * sqrtf(0.25f * ss / 127.0f);
        *out_loss = -rec + dkl + ldv + lsm;
    }
}

extern "C" void kernel_launch(void* const* d_in, const int* in_sizes, int n_in,
                              void* d_out, int out_size, void* d_ws, size_t ws_size,
                              hipStream_t stream) {
    (void)in_sizes; (void)n_in; (void)out_size; (void)ws_size;
    const float* x   = (const float*)d_in[0];   // 8192 x 784
    const float* eps = (const float*)d_in[1];   // 8192 x 128
    const float* W11 = (const float*)d_in[2];   // 784 x 3136
    const float* b11 = (const float*)d_in[3];
    const float* W31 = (const float*)d_in[4];   // 3136 x 128
    const float* b31 = (const float*)d_in[5];
    const float* W32 = (const float*)d_in[6];   // 3136 x 128
    const float* b32 = (const float*)d_in[7];
    const float* W44 = (const float*)d_in[8];   // 128 x 392
    const float* b44 = (const float*)d_in[9];
    const float* W61 = (const float*)d_in[10];  // 392 x 784
    const float* b61 = (const float*)d_in[11];

    const int M = 8192, IM = 784, H = 3136, L = 128, D2 = 392;
    const int SPLITK = 4, KS = H / SPLITK;      // 3136/4 = 784 per slice

    float* ws       = (float*)d_ws;
    float* enc      = ws;                                  // 8192*3136
    float* mu       = enc + (size_t)M * H;                 // 8192*128
    float* sig      = mu + (size_t)M * L;                  // 8192*128
    float* z        = sig + (size_t)M * L;                 // 8192*128
    float* dec1     = z + (size_t)M * L;                   // 8192*392
    float* dquad    = dec1 + (size_t)M * D2;               // 4096
    float* tvec     = dquad + 4096;                        // 4096
    float* dklrow   = tvec + 4096;                         // 4096
    float* recpart  = dklrow + 4096;                       // 1024
    float* ldivpart = recpart + 1024;                      // 16
    float* skpart   = ldivpart + 16;                       // 4*8192*128

    float* y        = (float*)d_out;                       // 8192*784
    float* out_loss = y + (size_t)M * IM;                  // scalar

    dim3 blk(256);
    // enc = relu(x @ W11 + b11)
    vae_gemm_bias_act<<<dim3((H + 127) / 128, M / 128), blk, 0, stream>>>(
        x, W11, b11, enc, M, H, IM, 0, IM, 0);
    // mu = relu(enc @ W31 + b31) via split-K=4 (deterministic fixed-order combine)
    for (int s = 0; s < SPLITK; s++)
        vae_gemm_bias_act<<<dim3(1, M / 128), blk, 0, stream>>>(
            enc, W31, b31, skpart + (size_t)s * M * L, M, L, H, s * KS, (s + 1) * KS, 2);
    vae_combine_relu<<<dim3((M * L + 255) / 256), blk, 0, stream>>>(
        skpart, b31, mu, M * L, L, SPLITK);
    // sig = relu(enc @ W32 + b32) via split-K=4
    for (int s = 0; s < SPLITK; s++)
        vae_gemm_bias_act<<<dim3(1, M / 128), blk, 0, stream>>>(
            enc, W32, b32, skpart + (size_t)s * M * L, M, L, H, s * KS, (s + 1) * KS, 2);
    vae_combine_relu<<<dim3((M * L + 255) / 256), blk, 0, stream>>>(
        skpart, b32, sig, M * L, L, SPLITK);
    // z = eps * sig + mu
    vae_reparam<<<dim3((M * L + 255) / 256), blk, 0, stream>>>(eps, sig, mu, z, M * L);
    // dec1 = relu(z @ W44 + b44)
    vae_gemm_bias_act<<<dim3((D2 + 127) / 128, M / 128), blk, 0, stream>>>(
        z, W44, b44, dec1, M, D2, L, 0, L, 0);
    // y = sigmoid(relu(dec1 @ W61 + b61))
    vae_gemm_bias_act<<<dim3((IM + 127) / 128, M / 128), blk, 0, stream>>>(
        dec1, W61, b61, y, M, IM, D2, 0, D2, 1);
    // per-row KL / trace / dquad / t
    vae_stats<<<dim3(4096), dim3(128), 0, stream>>>(mu, sig, dquad, tvec, dklrow);
    // pairwise softplus partials
    vae_ldiverse<<<dim3(16), blk, 0, stream>>>(dquad, tvec, ldivpart);
    // reconstruction loss partials
    vae_rec<<<dim3(1024), blk, 0, stream>>>(x, y, recpart, M * IM);
    // deterministic final combine
    vae_final<<<dim3(1), dim3(32), 0, stream>>>(dklrow, recpart, ldivpart, tvec, out_loss);
}